// AlltoAllTokenDispatcher_50002009260703
// MI455X (gfx1250) — compile-verified
//
#include <hip/hip_runtime.h>
#include <hip/hip_bf16.h>
#include <stdint.h>

// Reference reduces algebraically to: out[n,:] = (probs[n,0]+probs[n,1]) * tokens[n,:]
// (gather->stable-sort->scatter_add is an exact identity permutation; `indices`
// do not affect the output). Pure HBM-bandwidth-bound streaming op:
// ~537 MB of traffic -> ~23 us floor at 23.3 TB/s on MI455X.
//
// We route the row through LDS with the CDNA5 async data mover
// (GLOBAL_LOAD_ASYNC_TO_LDS_B128 / ASYNCcnt) — free at the HBM roofline,
// and it keeps a deep memory pipeline without VGPR return dependencies.

#define N_TOKENS 16384
#define HIDDEN   4096
#define TOPK     2
#define THREADS  256
#define VEC_PER_THREAD 4   // 4 x float4 = 16 floats; 256*16 = 4096 = HIDDEN

// Builtin signature (from hipcc diagnostic):
//   __builtin_amdgcn_global_load_async_to_lds_b128(int4 AS1* src, int4 AS3* dst,
//                                                  imm offset, imm cpol)
typedef int v4i __attribute__((vector_size(16)));
typedef __attribute__((address_space(1))) v4i g_v4i;   // global int4
typedef __attribute__((address_space(3))) v4i l_v4i;   // LDS int4

__device__ __forceinline__ g_v4i* to_global_v4(const void* p) {
  // generic global pointer value == addrspace(1) value on amdgcn
  return (g_v4i*)(uintptr_t)p;
}
__device__ __forceinline__ l_v4i* to_lds_v4(const void* p) {
  // low 32 bits of a generic LDS address are the LDS byte offset (aperture map)
  return (l_v4i*)(uint32_t)(uintptr_t)p;
}

__global__ void __launch_bounds__(THREADS)
moe_dispatch_combine_kernel(const float* __restrict__ tokens,
                            const float* __restrict__ probs,
                            float* __restrict__ out) {
  const int row = blockIdx.x;
  const int t   = threadIdx.x;

  // Uniform per block -> scalar loads.
  const float scale = probs[2 * row] + probs[2 * row + 1];

  const float4* __restrict__ src = (const float4*)(tokens + (size_t)row * HIDDEN);
  float4* __restrict__ dst       = (float4*)(out    + (size_t)row * HIDDEN);

#if defined(__gfx1250__) && __has_builtin(__builtin_amdgcn_global_load_async_to_lds_b128)
  __shared__ __align__(16) float lbuf[HIDDEN];  // 16 KB of the 320 KB/WGP pool

  // Stage the whole row into LDS via the async data-mover path.
  // Each lane's b128 lands at the matching LDS offset (identity mapping),
  // so each lane reads back exactly the bytes it requested.
#pragma unroll
  for (int j = 0; j < VEC_PER_THREAD; ++j) {
    const int i = t + THREADS * j;           // float4 index within the row
    __builtin_amdgcn_global_load_async_to_lds_b128(
        to_global_v4(src + i), to_lds_v4(lbuf + 4 * i), /*offset=*/0, /*cpol=*/0);
  }

#if __has_builtin(__builtin_amdgcn_s_wait_asynccnt)
  __builtin_amdgcn_s_wait_asynccnt(0);
#else
  asm volatile("s_wait_asynccnt 0x0" ::: "memory");
#endif
  __syncthreads();  // cheap insurance; wave-local wait already covers own lanes

#pragma unroll
  for (int j = 0; j < VEC_PER_THREAD; ++j) {
    const int i = t + THREADS * j;
    float4 v = *(const float4*)(lbuf + 4 * i);  // ds_load_b128
    v.x *= scale; v.y *= scale; v.z *= scale; v.w *= scale;
    dst[i] = v;                                  // global_store_b128
  }
#else
  // Fallback (host pass / toolchains without the async builtins):
  // plain B128 streaming, still roofline-optimal.
#pragma unroll
  for (int j = 0; j < VEC_PER_THREAD; ++j) {
    const int i = t + THREADS * j;
    float4 v = src[i];
    v.x *= scale; v.y *= scale; v.z *= scale; v.w *= scale;
    dst[i] = v;
  }
#endif
}

extern "C" void kernel_launch(void* const* d_in, const int* in_sizes, int n_in,
                              void* d_out, int out_size, void* d_ws, size_t ws_size,
                              hipStream_t stream) {
  (void)in_sizes; (void)n_in; (void)out_size; (void)d_ws; (void)ws_size;
  const float* tokens = (const float*)d_in[0];
  const float* probs  = (const float*)d_in[1];
  // d_in[2] (expert indices) is provably irrelevant to the output — the
  // gather/sort/scatter round trip is the identity permutation.
  float* out = (float*)d_out;

  dim3 grid(N_TOKENS);
  dim3 block(THREADS);
  hipLaunchKernelGGL(moe_dispatch_combine_kernel, grid, block, 0, stream,
                     tokens, probs, out);
}